// HybridSSIMRMIFuse_12489764897408
// MI455X (gfx1250) — compile-verified
//
#include <hip/hip_runtime.h>
#include <math.h>

typedef __attribute__((ext_vector_type(2))) float v2f;
typedef __attribute__((ext_vector_type(8))) float v8f;

#define BATCH 32
#define H 512
#define W 512
#define PH 171                 // pooled size: floor((512+2-3)/3)+1
#define PO 169                 // patch grid: PH - 2
#define LCOLS (PO*PO)          // 28561 samples per gram
#define NPIX ((double)BATCH*H*W)

#define TS 32                  // ssim output tile
#define TIN (TS+10)            // 42: tile + 2*5 halo

// workspace layout (bytes)
#define ACC_OFF   0                              // 4 doubles: [0]=ssim pair0 sum, [1]=ssim pair1 sum, [2]=bce sum
#define GRAM_OFF  256                            // 64 * 3 * 256 floats
#define SUMS_OFF  (GRAM_OFF + 64*3*256*4)        // 64 * 32 floats (row sums y, p)
#define POOL_OFF  (SUMS_OFF + 64*32*4)           // 3 maps * 32 * 171 * 171 floats
#define ZERO_WORDS (POOL_OFF/4)

// normalized 1D Gaussian, WIN=11, SIGMA=1.5
__constant__ float GW[11] = {
    0.00102838f, 0.00759878f, 0.03600077f, 0.10936069f, 0.21300553f,
    0.26601172f,
    0.21300553f, 0.10936069f, 0.03600077f, 0.00759878f, 0.00102838f
};

__global__ void zero_ws_kernel(unsigned int* w, int n) {
    int i = blockIdx.x * blockDim.x + threadIdx.x;
    if (i < n) w[i] = 0u;
}

// ---------------- SSIM: separable 11x11 Gaussian, 32x32 tile per block ----------------
// Stage 1: horizontal conv of {f, f^2, t, t^2, f*t} for 42 rows x 32 cols -> LDS
// Stage 2: vertical conv, SSIM map value, block reduction -> f64 atomic
__global__ __launch_bounds__(256) void ssim_kernel(const float* __restrict__ fuse,
                                                   const float* __restrict__ x,
                                                   double* __restrict__ acc) {
    const int tilesPerImg = (H/TS)*(W/TS);            // 256
    int blk  = blockIdx.x;
    int pair = blk / (BATCH * tilesPerImg);
    int rem  = blk % (BATCH * tilesPerImg);
    int b    = rem / tilesPerImg;
    int tile = rem % tilesPerImg;
    int ty0  = (tile / (W/TS)) * TS;
    int tx0  = (tile % (W/TS)) * TS;

    __shared__ float sf[TIN*TIN];
    __shared__ float st[TIN*TIN];
    __shared__ float h1[TIN*TS];   // conv_h(f)
    __shared__ float h2[TIN*TS];   // conv_h(t)
    __shared__ float h3[TIN*TS];   // conv_h(f^2)
    __shared__ float h4[TIN*TS];   // conv_h(t^2)
    __shared__ float h5[TIN*TS];   // conv_h(f*t)
    __shared__ double red[256];

    const float* fb = fuse + (size_t)b * H * W;
    const float* tb = x + ((size_t)b * 2 + pair) * H * W;

    // load halo'd inputs (zero padding at image borders)
    for (int i = threadIdx.x; i < TIN*TIN; i += 256) {
        int r = i / TIN, c = i % TIN;
        int gy = ty0 + r - 5, gx = tx0 + c - 5;
        float fv = 0.f, tv = 0.f;
        if (gy >= 0 && gy < H && gx >= 0 && gx < W) {
            fv = fb[gy*W + gx];
            fv = fminf(fmaxf(fv, 0.f), 1.f);      // fuse clipped to [0,1]
            tv = tb[gy*W + gx];
        }
        sf[i] = fv; st[i] = tv;
    }
    __syncthreads();

    // stage 1: horizontal taps over 42 rows x 32 output columns
    for (int i = threadIdx.x; i < TIN*TS; i += 256) {
        int r = i / TS, xx = i % TS;
        int base = r * TIN + xx;
        float a1 = 0.f, a2 = 0.f, a3 = 0.f, a4 = 0.f, a5 = 0.f;
        #pragma unroll
        for (int j = 0; j < 11; ++j) {
            float g = GW[j];
            float f = sf[base + j], t = st[base + j];
            a1 += g * f;     a2 += g * t;
            a3 += g * f * f; a4 += g * t * t; a5 += g * f * t;
        }
        h1[i] = a1; h2[i] = a2; h3[i] = a3; h4[i] = a4; h5[i] = a5;
    }
    __syncthreads();

    // stage 2: vertical taps, 4 output pixels per thread
    const float C1v = 1e-4f, C2v = 9e-4f;
    double local = 0.0;
    #pragma unroll
    for (int k = 0; k < 4; ++k) {
        int idx = threadIdx.x + k * 256;      // 0..1023
        int y = idx / TS, xx = idx % TS;
        float mu1 = 0.f, mu2 = 0.f, s11 = 0.f, s22 = 0.f, s12 = 0.f;
        #pragma unroll
        for (int i = 0; i < 11; ++i) {
            float g = GW[i];
            int o = (y + i) * TS + xx;
            mu1 += g * h1[o]; mu2 += g * h2[o];
            s11 += g * h3[o]; s22 += g * h4[o]; s12 += g * h5[o];
        }
        float mu1s = mu1*mu1, mu2s = mu2*mu2, m12 = mu1*mu2;
        float sig1 = s11 - mu1s, sig2 = s22 - mu2s, sig12 = s12 - m12;
        float ssim = ((2.f*m12 + C1v) * (2.f*sig12 + C2v)) /
                     ((mu1s + mu2s + C1v) * (sig1 + sig2 + C2v));
        local += (double)ssim;
    }

    red[threadIdx.x] = local;
    __syncthreads();
    for (int s = 128; s > 0; s >>= 1) {
        if (threadIdx.x < s) red[threadIdx.x] += red[threadIdx.x + s];
        __syncthreads();
    }
    if (threadIdx.x == 0) atomicAdd(&acc[pair], red[0]);
}

// ---------------- BCE over both pairs ----------------
__global__ __launch_bounds__(256) void bce_kernel(const float* __restrict__ fuse,
                                                  const float* __restrict__ x,
                                                  double* __restrict__ acc) {
    size_t idx    = (size_t)blockIdx.x * blockDim.x + threadIdx.x;
    size_t stride = (size_t)gridDim.x * blockDim.x;
    size_t n = (size_t)BATCH * H * W;
    double local = 0.0;
    for (size_t i = idx; i < n; i += stride) {
        size_t b = i / (H*W), hw = i % (H*W);
        float p = fuse[i];
        p = fminf(fmaxf(p, 0.f), 1.f);
        p = fminf(fmaxf(p, 1e-12f), 1.f - 1e-12f);
        float lp  = logf(p);
        float l1p = logf(1.f - p);
        float tv = x[(b*2 + 0)*(size_t)(H*W) + hw];
        float ti = x[(b*2 + 1)*(size_t)(H*W) + hw];
        local += (double)(-(tv*lp + (1.f - tv)*l1p));
        local += (double)(-(ti*lp + (1.f - ti)*l1p));
    }
    __shared__ double red[256];
    red[threadIdx.x] = local;
    __syncthreads();
    for (int s = 128; s > 0; s >>= 1) {
        if (threadIdx.x < s) red[threadIdx.x] += red[threadIdx.x + s];
        __syncthreads();
    }
    if (threadIdx.x == 0) atomicAdd(&acc[2], red[0]);
}

// ---------------- 3x3 stride-3 max pool, pad 1; maps: 0=fuse(clipped), 1=vis, 2=ir ----------
__global__ __launch_bounds__(256) void pool_kernel(const float* __restrict__ fuse,
                                                   const float* __restrict__ x,
                                                   float* __restrict__ pool) {
    int total = 3 * BATCH * PH * PH;
    int i = blockIdx.x * blockDim.x + threadIdx.x;
    if (i >= total) return;
    int m = i / (BATCH*PH*PH);
    int r = i % (BATCH*PH*PH);
    int b = r / (PH*PH);
    int q = r % (PH*PH);
    int oy = q / PH, ox = q % PH;
    const float* src = (m == 0) ? (fuse + (size_t)b*H*W)
                                : (x + ((size_t)b*2 + (m-1))*H*W);
    float best = -1e30f;
    int y0 = oy*3 - 1, x0 = ox*3 - 1;
    for (int dy = 0; dy < 3; ++dy) {
        int yy = y0 + dy; if (yy < 0 || yy >= H) continue;
        for (int dx = 0; dx < 3; ++dx) {
            int xx = x0 + dx; if (xx < 0 || xx >= W) continue;
            float v = src[yy*W + xx];
            if (m == 0) v = fminf(fmaxf(v, 0.f), 1.f);
            best = fmaxf(best, v);
        }
    }
    pool[i] = best;
}

// ---------------- Gram kernel: V_WMMA_F32_16X16X4_F32 accumulates Syy, Spp, Syp ----------
// block = one (b,pair); 8 waves split the 28561 columns; rows 0..8 = 3x3 patch offsets, 9..15 = 0 pad.
// Column index marches by +32 per wave-iteration -> incremental (oy,ox) tracking, no division.
__global__ __launch_bounds__(256) void gram_kernel(const float* __restrict__ pool,
                                                   float* __restrict__ grams,
                                                   float* __restrict__ sums) {
    int bp   = blockIdx.x;          // 0..63
    int b    = bp >> 1;
    int pair = bp & 1;
    const float* pm = pool + (size_t)b * PH * PH;                           // pooled fuse
    const float* ym = pool + (size_t)(1 + pair) * BATCH * PH * PH
                           + (size_t)b * PH * PH;                           // pooled target

    int lane = threadIdx.x & 31;
    int wave = threadIdx.x >> 5;    // 0..7
    int m    = lane & 15;           // row index in 16x16 tile
    int kk   = (lane >> 4) << 1;    // K sub-offset: lanes 0-15 -> k0,k1 ; 16-31 -> k2,k3
    int di   = m / 3, dj = m % 3;
    bool rowvalid = (m < 9);

    v8f ayy = {}; v8f app = {}; v8f ayp = {};
    float sy = 0.f, sp = 0.f;

    // starting column for this lane
    int l  = wave * 4 + kk;
    int oy = l / PO;                // tiny one-time division
    int ox = l - oy * PO;

    const int nchunks = (LCOLS + 3) / 4;   // 7141
    for (int c = wave; c < nchunks; c += 8) {
        v2f yv = {0.f, 0.f};
        v2f pv = {0.f, 0.f};
        if (rowvalid) {
            if (l < LCOLS) {
                int off = (oy + di) * PH + (ox + dj);
                yv[0] = ym[off];
                pv[0] = pm[off];
            }
            // second column (l+1): ox+1 wraps at most once
            int oy1 = oy, ox1 = ox + 1;
            if (ox1 == PO) { ox1 = 0; oy1 += 1; }
            if (l + 1 < LCOLS) {
                int off1 = (oy1 + di) * PH + (ox1 + dj);
                yv[1] = ym[off1];
                pv[1] = pm[off1];
            }
        }
        sy += yv[0] + yv[1];
        sp += pv[0] + pv[1];
        // Gram: A-tile and B-tile lane contents coincide, so pass the same regs.
        ayy = __builtin_amdgcn_wmma_f32_16x16x4_f32(false, yv, false, yv, (short)0, ayy, false, false);
        app = __builtin_amdgcn_wmma_f32_16x16x4_f32(false, pv, false, pv, (short)0, app, false, false);
        ayp = __builtin_amdgcn_wmma_f32_16x16x4_f32(false, yv, false, pv, (short)0, ayp, false, false);

        // advance 32 columns (8 waves * 4 cols); 32 < PO so at most one wrap
        l  += 32;
        ox += 32;
        if (ox >= PO) { ox -= PO; oy += 1; }
    }

    // D layout: VGPR r, lanes 0-15 -> (M=r, N=lane); lanes 16-31 -> (M=r+8, N=lane-16)
    float* g = grams + (size_t)bp * 768;
    int row0 = (lane >> 4) * 8;
    int col  = lane & 15;
    #pragma unroll
    for (int r = 0; r < 8; ++r) {
        int row = row0 + r;
        atomicAdd(&g[0*256 + row*16 + col], ayy[r]);
        atomicAdd(&g[1*256 + row*16 + col], app[r]);
        atomicAdd(&g[2*256 + row*16 + col], ayp[r]);
    }
    float* s = sums + bp * 32;
    atomicAdd(&s[m],      sy);
    atomicAdd(&s[16 + m], sp);
}

// ---------------- finalize: f64 9x9 inverse + Cholesky per (b,pair), combine scalars ----------
__global__ __launch_bounds__(64) void finalize_kernel(const float* __restrict__ grams,
                                                      const float* __restrict__ sums,
                                                      const double* __restrict__ acc,
                                                      float* __restrict__ out) {
    __shared__ double rmish[64];
    int bp = threadIdx.x;

    const float* g = grams + (size_t)bp * 768;
    const float* s = sums + bp * 32;
    const double invL = 1.0 / (double)LCOLS;
    const double EPS = 0.0005;

    double sy[9], sp[9];
    for (int i = 0; i < 9; ++i) { sy[i] = (double)s[i]; sp[i] = (double)s[16 + i]; }

    double yc[9][9], pc[9][9], ypc[9][9];
    for (int i = 0; i < 9; ++i)
        for (int j = 0; j < 9; ++j) {
            yc[i][j]  = (double)g[        i*16 + j] - sy[i]*sy[j]*invL;
            pc[i][j]  = (double)g[256  +  i*16 + j] - sp[i]*sp[j]*invL;
            ypc[i][j] = (double)g[512  +  i*16 + j] - sy[i]*sp[j]*invL;
        }

    // Gauss-Jordan inverse of (pc + EPS*I)  (SPD, no pivoting)
    double a[9][18];
    for (int i = 0; i < 9; ++i)
        for (int j = 0; j < 9; ++j) {
            a[i][j]     = pc[i][j] + (i == j ? EPS : 0.0);
            a[i][9 + j] = (i == j) ? 1.0 : 0.0;
        }
    for (int k = 0; k < 9; ++k) {
        double ip = 1.0 / a[k][k];
        for (int j = 0; j < 18; ++j) a[k][j] *= ip;
        for (int i = 0; i < 9; ++i) {
            if (i == k) continue;
            double f = a[i][k];
            for (int j = 0; j < 18; ++j) a[i][j] -= f * a[k][j];
        }
    }
    // inv[i][j] = a[i][9+j];  p_inv_t[i][j] = inv[j][i]
    // t = ypc @ p_inv_t : t[i][j] = sum_k ypc[i][k] * inv[j][k]
    double t[9][9];
    for (int i = 0; i < 9; ++i)
        for (int j = 0; j < 9; ++j) {
            double acc2 = 0.0;
            for (int k = 0; k < 9; ++k) acc2 += ypc[i][k] * a[j][9 + k];
            t[i][j] = acc2;
        }
    // m = yc - t @ ypc^T
    double mm[9][9];
    for (int i = 0; i < 9; ++i)
        for (int j = 0; j < 9; ++j) {
            double acc2 = 0.0;
            for (int k = 0; k < 9; ++k) acc2 += t[i][k] * ypc[j][k];
            mm[i][j] = yc[i][j] - acc2;
        }
    // Cholesky of (mm + EPS*I)
    double ch[9][9];
    for (int j = 0; j < 9; ++j) {
        double d = mm[j][j] + EPS;
        for (int k = 0; k < j; ++k) d -= ch[j][k] * ch[j][k];
        ch[j][j] = sqrt(d);
        double inv_d = 1.0 / ch[j][j];
        for (int i = j + 1; i < 9; ++i) {
            double v = mm[i][j];
            for (int k = 0; k < j; ++k) v -= ch[i][k] * ch[j][k];
            ch[i][j] = v * inv_d;
        }
    }
    double slog = 0.0;
    for (int j = 0; j < 9; ++j) slog += log(ch[j][j] + 1e-8);
    rmish[bp] = slog / 9.0;
    __syncthreads();

    if (bp == 0) {
        double rsum = 0.0;
        for (int i = 0; i < 64; ++i) rsum += rmish[i];
        double ssim_loss = (1.0 - acc[0] / NPIX) + (1.0 - acc[1] / NPIX);
        double rmi_loss  = (rsum / (double)BATCH) * 0.4 + (acc[2] / NPIX) * 0.6;
        out[0] = (float)(ssim_loss + rmi_loss);
        out[1] = (float)ssim_loss;
        out[2] = (float)rmi_loss;
    }
}

extern "C" void kernel_launch(void* const* d_in, const int* in_sizes, int n_in,
                              void* d_out, int out_size, void* d_ws, size_t ws_size,
                              hipStream_t stream) {
    const float* fuse = (const float*)d_in[0];   // (32,1,512,512) f32
    const float* x    = (const float*)d_in[1];   // (32,2,512,512) f32
    float* out = (float*)d_out;

    char* ws = (char*)d_ws;
    double* acc  = (double*)(ws + ACC_OFF);
    float* grams = (float*)(ws + GRAM_OFF);
    float* sums  = (float*)(ws + SUMS_OFF);
    float* pool  = (float*)(ws + POOL_OFF);

    zero_ws_kernel<<<(ZERO_WORDS + 255)/256, 256, 0, stream>>>((unsigned int*)ws, ZERO_WORDS);

    int ssim_blocks = 2 * BATCH * (H/TS) * (W/TS);               // 16384
    ssim_kernel<<<ssim_blocks, 256, 0, stream>>>(fuse, x, acc);

    bce_kernel<<<2048, 256, 0, stream>>>(fuse, x, acc);

    int pool_total = 3 * BATCH * PH * PH;
    pool_kernel<<<(pool_total + 255)/256, 256, 0, stream>>>(fuse, x, pool);

    gram_kernel<<<64, 256, 0, stream>>>(pool, grams, sums);

    finalize_kernel<<<1, 64, 0, stream>>>(grams, sums, acc, out);
}